// GNNEncoder_80479097192490
// MI455X (gfx1250) — compile-verified
//
#include <hip/hip_runtime.h>
#include <math.h>

#define IN_DIM 128
#define HID 64
#define HEADS 4
#define NEG_SLOPE 0.2f

typedef __attribute__((ext_vector_type(2))) float v2f;
typedef __attribute__((ext_vector_type(8))) float v8f;

// ---------------------------------------------------------------------------
// Dense GEMM D[N,M] = A[N,K] @ B[K,M] using V_WMMA_F32_16X16X4_F32 (fp32).
// One wave (32 lanes) computes a 16x64 output strip (4 accumulators), looping
// K in steps of 4 and reusing the A fragment across 4 column tiles.
// A-layout (16x4, MxK): lanes 0-15 hold row=lane, v0=K0,v1=K1; lanes 16-31 K2,K3.
// B-layout (4x16, KxN): lanes 0-15 hold col=lane, v0=K0,v1=K1; lanes 16-31 K2,K3.
// C/D-layout: vgpr r, lanes 0-15 -> row r; lanes 16-31 -> row 8+r; col = lane&15.
// Requires N%16==0, M%64==0, K%4==0 (true here: N=50000=16*3125, M in {256,64}).
// ---------------------------------------------------------------------------
__global__ __launch_bounds__(32) void gemm_wmma_16x64(
    const float* __restrict__ A, const float* __restrict__ B,
    float* __restrict__ D, int K, int M) {
  const int lane = threadIdx.x;
  const int half = lane >> 4;       // which K-half this lane holds
  const int l    = lane & 15;
  const size_t row = (size_t)blockIdx.x * 16 + l;
  const int col0   = blockIdx.y * 64;
  const float* aptr  = A + row * (size_t)K + half * 2;
  const float* bbase = B + (size_t)(half * 2) * M + col0 + l;
  v8f acc0 = {}, acc1 = {}, acc2 = {}, acc3 = {};
#pragma unroll 2
  for (int k = 0; k < K; k += 4) {
    v2f a;
    a.x = aptr[k];
    a.y = aptr[k + 1];
    const float* bk = bbase + (size_t)k * M;
    v2f b0, b1, b2, b3;
    b0.x = bk[0];  b0.y = bk[(size_t)M + 0];
    b1.x = bk[16]; b1.y = bk[(size_t)M + 16];
    b2.x = bk[32]; b2.y = bk[(size_t)M + 32];
    b3.x = bk[48]; b3.y = bk[(size_t)M + 48];
    acc0 = __builtin_amdgcn_wmma_f32_16x16x4_f32(false, a, false, b0, (short)0, acc0, false, false);
    acc1 = __builtin_amdgcn_wmma_f32_16x16x4_f32(false, a, false, b1, (short)0, acc1, false, false);
    acc2 = __builtin_amdgcn_wmma_f32_16x16x4_f32(false, a, false, b2, (short)0, acc2, false, false);
    acc3 = __builtin_amdgcn_wmma_f32_16x16x4_f32(false, a, false, b3, (short)0, acc3, false, false);
  }
  float* dptr = D + ((size_t)blockIdx.x * 16 + half * 8) * (size_t)M + col0 + l;
#pragma unroll
  for (int r = 0; r < 8; ++r) {
    dptr[(size_t)r * M + 0]  = acc0[r];
    dptr[(size_t)r * M + 16] = acc1[r];
    dptr[(size_t)r * M + 32] = acc2[r];
    dptr[(size_t)r * M + 48] = acc3[r];
  }
}

// ---------------------------------------------------------------------------
// Per-node attention logits: alpha_s[n,h] = sum_c h[n,h,c]*a_src[h,c] (and a_dst).
// One block per node, blockDim.x == F == H*64; LDS tree-reduce per 64-channel head.
// ---------------------------------------------------------------------------
__global__ void alpha_kernel(const float* __restrict__ h,
                             const float* __restrict__ a_s,
                             const float* __restrict__ a_d,
                             float* __restrict__ out_s,
                             float* __restrict__ out_d) {
  __shared__ float s0[256];
  __shared__ float s1[256];
  const int F = blockDim.x;
  const int t = threadIdx.x;
  const size_t nid = blockIdx.x;
  float hv = h[nid * (size_t)F + t];
  s0[t] = hv * a_s[t];
  s1[t] = hv * a_d[t];
  __syncthreads();
  for (int off = 32; off >= 1; off >>= 1) {
    if ((t & 63) < off) {
      s0[t] += s0[t + off];
      s1[t] += s1[t + off];
    }
    __syncthreads();
  }
  if ((t & 63) == 0) {
    int head = t >> 6;
    int H = F >> 6;
    out_s[nid * H + head] = s0[t];
    out_d[nid * H + head] = s1[t];
  }
}

__global__ void init_softmax_kernel(float* __restrict__ m, float* __restrict__ denom,
                                    size_t total) {
  size_t i = (size_t)blockIdx.x * blockDim.x + threadIdx.x;
  if (i < total) {
    m[i] = -__builtin_inff();
    denom[i] = 0.0f;
  }
}

__global__ void init_bias_kernel(float* __restrict__ out, const float* __restrict__ bias,
                                 int F, size_t total) {
  size_t i = (size_t)blockIdx.x * blockDim.x + threadIdx.x;
  if (i < total) out[i] = bias[i & (size_t)(F - 1)];  // F is a power of two
}

__device__ __forceinline__ void edge_nodes(const int* __restrict__ src,
                                           const int* __restrict__ dst,
                                           int E, int e, int& s, int& d) {
  if (e < E) { s = src[e]; d = dst[e]; }
  else       { s = e - E; d = e - E; }   // implicit self-loop
}

__device__ __forceinline__ float leaky(float v) {
  return v > 0.0f ? v : v * NEG_SLOPE;
}

// float atomic max via signed/unsigned integer atomics (valid incl. negatives,
// target initialized to -inf).
__device__ __forceinline__ void atomic_max_f32(float* addr, float val) {
  if (val >= 0.0f)
    atomicMax((int*)addr, __float_as_int(val));
  else
    atomicMin((unsigned int*)addr, __float_as_uint(val));
}

// Pass 1: per-(edge,head) logit -> segment max over dst.
__global__ void edge_max_kernel(const int* __restrict__ src, const int* __restrict__ dst,
                                int E, int ET, int H,
                                const float* __restrict__ as_, const float* __restrict__ ad_,
                                float* __restrict__ m) {
  size_t i = (size_t)blockIdx.x * blockDim.x + threadIdx.x;
  if (i >= (size_t)ET * H) return;
  int e = (int)(i / H);
  int hh = (int)(i % H);
  int s, d;
  edge_nodes(src, dst, E, e, s, d);
  float ev = leaky(as_[(size_t)s * H + hh] + ad_[(size_t)d * H + hh]);
  atomic_max_f32(&m[(size_t)d * H + hh], ev);
}

// Pass 2: ex = exp(e - m[dst]); segment-sum into denom; stash ex per edge.
__global__ void edge_exp_kernel(const int* __restrict__ src, const int* __restrict__ dst,
                                int E, int ET, int H,
                                const float* __restrict__ as_, const float* __restrict__ ad_,
                                const float* __restrict__ m,
                                float* __restrict__ ex, float* __restrict__ denom) {
  size_t i = (size_t)blockIdx.x * blockDim.x + threadIdx.x;
  if (i >= (size_t)ET * H) return;
  int e = (int)(i / H);
  int hh = (int)(i % H);
  int s, d;
  edge_nodes(src, dst, E, e, s, d);
  float ev = leaky(as_[(size_t)s * H + hh] + ad_[(size_t)d * H + hh]);
  float xv = __expf(ev - m[(size_t)d * H + hh]);
  ex[(size_t)e * H + hh] = xv;
  atomicAdd(&denom[(size_t)d * H + hh], xv);
}

// Pass 3: out[dst, c..c+3] += (ex/denom) * h[src, c..c+3]
// One thread per edge-float4: b128 gathers of h[src] (L2-resident), f32 atomic
// scatters into out[dst]. A float4 never crosses a 64-channel head boundary.
__global__ void edge_agg_kernel(const int* __restrict__ src, const int* __restrict__ dst,
                                int E, int ET, int H, int lgFq,  // lg(F/4)
                                const float* __restrict__ h,
                                const float* __restrict__ ex, const float* __restrict__ denom,
                                float* __restrict__ out) {
  size_t i = (size_t)blockIdx.x * blockDim.x + threadIdx.x;
  if (i >= ((size_t)ET << lgFq)) return;
  const int F = 4 << lgFq;
  int e = (int)(i >> lgFq);
  int c = ((int)(i & (size_t)((1 << lgFq) - 1))) << 2;  // channel base (multiple of 4)
  int hh = c >> 6;                                      // 64 channels per head
  int s, d;
  edge_nodes(src, dst, E, e, s, d);
  float alpha = ex[(size_t)e * H + hh] / denom[(size_t)d * H + hh];
  const float4 hv = *(const float4*)(h + (size_t)s * F + c);
  float* op = out + (size_t)d * F + c;
  atomicAdd(op + 0, alpha * hv.x);
  atomicAdd(op + 1, alpha * hv.y);
  atomicAdd(op + 2, alpha * hv.z);
  atomicAdd(op + 3, alpha * hv.w);
}

__global__ void elu_kernel(float* __restrict__ x, size_t total) {
  size_t i = (size_t)blockIdx.x * blockDim.x + threadIdx.x;
  if (i < total) {
    float v = x[i];
    x[i] = v > 0.0f ? v : expm1f(v);
  }
}

// ---------------------------------------------------------------------------
// Host-side layer driver (all launches on `stream`, graph-capture safe).
// ---------------------------------------------------------------------------
static void run_layer(const float* in, int K, const float* W,
                      const float* a_s, const float* a_d, const float* bias,
                      int H, const int* srcE, const int* dstE, int N_, int E_,
                      float* hbuf, float* outbuf,
                      float* asb, float* adb, float* mb, float* db, float* exb,
                      bool do_elu, hipStream_t stream) {
  const int F = H * HID;
  const int ET = E_ + N_;
  const int lgFq = (F == 256) ? 6 : 4;  // lg(F/4)

  dim3 gGemm(N_ / 16, F / 64);
  gemm_wmma_16x64<<<gGemm, dim3(32), 0, stream>>>(in, W, hbuf, K, F);

  alpha_kernel<<<dim3(N_), dim3(F), 0, stream>>>(hbuf, a_s, a_d, asb, adb);

  size_t nh = (size_t)N_ * H;
  init_softmax_kernel<<<(unsigned)((nh + 255) / 256), 256, 0, stream>>>(mb, db, nh);

  size_t nf = (size_t)N_ * F;
  init_bias_kernel<<<(unsigned)((nf + 255) / 256), 256, 0, stream>>>(outbuf, bias, F, nf);

  size_t eh = (size_t)ET * H;
  edge_max_kernel<<<(unsigned)((eh + 255) / 256), 256, 0, stream>>>(
      srcE, dstE, E_, ET, H, asb, adb, mb);
  edge_exp_kernel<<<(unsigned)((eh + 255) / 256), 256, 0, stream>>>(
      srcE, dstE, E_, ET, H, asb, adb, mb, exb, db);

  size_t eq = (size_t)ET << lgFq;  // edge-float4 work items
  edge_agg_kernel<<<(unsigned)((eq + 255) / 256), 256, 0, stream>>>(
      srcE, dstE, E_, ET, H, lgFq, hbuf, exb, db, outbuf);

  if (do_elu)
    elu_kernel<<<(unsigned)((nf + 255) / 256), 256, 0, stream>>>(outbuf, nf);
}

extern "C" void kernel_launch(void* const* d_in, const int* in_sizes, int n_in,
                              void* d_out, int out_size, void* d_ws, size_t ws_size,
                              hipStream_t stream) {
  const float* x   = (const float*)d_in[0];
  const int*   ei  = (const int*)d_in[1];
  const float* W0  = (const float*)d_in[2];
  const float* as0 = (const float*)d_in[3];
  const float* ad0 = (const float*)d_in[4];
  const float* b0  = (const float*)d_in[5];
  const float* W1  = (const float*)d_in[6];
  const float* as1 = (const float*)d_in[7];
  const float* ad1 = (const float*)d_in[8];
  const float* b1  = (const float*)d_in[9];
  const float* W2  = (const float*)d_in[10];
  const float* as2 = (const float*)d_in[11];
  const float* ad2 = (const float*)d_in[12];
  const float* b2  = (const float*)d_in[13];

  const int N_ = in_sizes[0] / IN_DIM;  // 50000 (multiple of 16)
  const int E_ = in_sizes[1] / 2;       // 400000
  const int* srcE = ei;                 // edge_index[0]
  const int* dstE = ei + E_;            // edge_index[1]

  const int Fmax = HEADS * HID;         // 256
  float* buf0 = (float*)d_ws;                         // h (gemm out)   [N,256]
  float* buf1 = buf0 + (size_t)N_ * Fmax;             // aggregated out [N,256]
  float* asb  = buf1 + (size_t)N_ * Fmax;             // alpha_src [N,H]
  float* adb  = asb + (size_t)N_ * HEADS;             // alpha_dst [N,H]
  float* mb   = adb + (size_t)N_ * HEADS;             // segment max [N,H]
  float* db   = mb  + (size_t)N_ * HEADS;             // segment sum [N,H]
  float* exb  = db  + (size_t)N_ * HEADS;             // per-edge exp [ET,H]

  // Layer 0: x[N,128] -> buf1[N,256], ELU
  run_layer(x, IN_DIM, W0, as0, ad0, b0, HEADS, srcE, dstE, N_, E_,
            buf0, buf1, asb, adb, mb, db, exb, /*elu=*/true, stream);
  // Layer 1: buf1[N,256] -> buf1[N,256] (gemm into buf0 first), ELU
  run_layer(buf1, Fmax, W1, as1, ad1, b1, HEADS, srcE, dstE, N_, E_,
            buf0, buf1, asb, adb, mb, db, exb, /*elu=*/true, stream);
  // Layer 2: buf1[N,256] -> d_out[N,64], single head, no ELU
  run_layer(buf1, Fmax, W2, as2, ad2, b2, 1, srcE, dstE, N_, E_,
            buf0, (float*)d_out, asb, adb, mb, db, exb, /*elu=*/false, stream);
}